// DirectionalMaskGenerator_20907900797189
// MI455X (gfx1250) — compile-verified
//
#include <hip/hip_runtime.h>
#include <stdint.h>

// ---------------------------------------------------------------------------
// DirectionalMaskGenerator for MI455X (gfx1250).
//
// Sparse reformulation: out[n,p] = OR_a OR_{r in window(a,p)} peak[n,a,r].
// Window in r has width 2*3.0/delta_rho ~ 5.95 -> 5 or 6 integer r per (a,p).
// N=16 peak bits packed into a uint16 per (a,r); dilated table D5 gives the
// exact window OR with two probes: D5[r_lo] | pk[r_hi].
//
// CDNA5 paths used:
//  * V_WMMA_F32_16X16X4_F32 computes the window-center matrix
//      f[a,p] = (cos a/dr)*x_p + (sin a/dr)*y_p + 180
//    for a 16-angle x 16-pixel tile per instruction (the reference's matmul).
//  * tensor_load_to_lds (TDM) + s_wait_tensorcnt broadcasts the 271KB D5
//    table into each WGP's 320KB LDS.
// ---------------------------------------------------------------------------

#define A_DIM 360
#define A_PAD 368                 // 23 full 16-angle WMMA tiles; rows 360..367 all-zero
#define A_TILES (A_PAD / 16)      // 23
#define R_DIM 360
#define NMAPS 16
#define HIMG  256
#define WIMG  256
#define HW    (HIMG * WIMG)
#define RPAD  368                 // padded row; index q = r + 2; zeros outside r in [0,360)
#define MAP   (A_DIM * R_DIM)
#define TPAD  (A_PAD * RPAD)      // 135424 table entries

constexpr double DELTA_RHO_D = 2.0 * 181.01933598375618 / 359.0; // 2*sqrt(128^2+128^2)/(R-1)
constexpr float  INV_DR = (float)(1.0 / DELTA_RHO_D);
constexpr float  CW_F   = (float)(3.0 / DELTA_RHO_D);            // ~2.974820

// workspace layout (bytes)
#define WS_GMAX 0                    // 16 floats
#define WS_CS   64                   // 368 float2 (2944 B)
#define WS_PKB  3008                 // 368*368 u16 = 270848 B
#define WS_D5   273856               // 368*368 u16

// ---------------------------------------------------------------- gmax ------
__global__ void gmax_kernel(const float* __restrict__ in, float* __restrict__ gmax) {
    __shared__ float red[256];
    const int n = blockIdx.x;
    const float* m = in + n * MAP;
    float v = -1e30f;
    for (int i = threadIdx.x; i < MAP; i += 256) v = fmaxf(v, m[i]);
    red[threadIdx.x] = v;
    __syncthreads();
    for (int s = 128; s > 0; s >>= 1) {
        if (threadIdx.x < s) red[threadIdx.x] = fmaxf(red[threadIdx.x], red[threadIdx.x + s]);
        __syncthreads();
    }
    if (threadIdx.x == 0) gmax[n] = red[0];
}

// ---------------------------------------------------------------- trig ------
__global__ void trig_kernel(float2* __restrict__ cs) {
    int i = blockIdx.x * 256 + threadIdx.x;
    if (i < A_PAD) {
        if (i < A_DIM) {
            float th = (float)i * (float)(3.14159265358979323846 / 360.0);
            cs[i] = make_float2(cosf(th), sinf(th));
        } else {
            cs[i] = make_float2(0.0f, 0.0f);  // pad angles -> f = 180 -> zero table rows
        }
    }
}

// ------------------------------------------------------------ peak bits -----
__device__ __forceinline__ float nbval(const float* m, int aa, int rr) {
    return (aa >= 0 && aa < A_DIM && rr >= 0 && rr < R_DIM) ? m[aa * R_DIM + rr] : -1e30f;
}

__global__ void peaks_kernel(const float* __restrict__ in, const float* __restrict__ gmax,
                             uint16_t* __restrict__ pkb) {
    int t = blockIdx.x * 256 + threadIdx.x;
    if (t >= TPAD) return;
    int a = t / RPAD;
    int r = t - a * RPAD - 2;
    uint32_t bits = 0;
    if (a < A_DIM && r >= 0 && r < R_DIM) {
#pragma unroll
        for (int n = 0; n < NMAPS; ++n) {
            const float* m = in + n * MAP;
            float v = m[a * R_DIM + r];
            // 3x3 SAME max-pool local-max: v == pooled  <=>  no neighbor > v
            float nb = -1e30f;
            nb = fmaxf(nb, nbval(m, a - 1, r - 1));
            nb = fmaxf(nb, nbval(m, a - 1, r    ));
            nb = fmaxf(nb, nbval(m, a - 1, r + 1));
            nb = fmaxf(nb, nbval(m, a    , r - 1));
            nb = fmaxf(nb, nbval(m, a    , r + 1));
            nb = fmaxf(nb, nbval(m, a + 1, r - 1));
            nb = fmaxf(nb, nbval(m, a + 1, r    ));
            nb = fmaxf(nb, nbval(m, a + 1, r + 1));
            if ((v >= nb) && (v > 0.5f * gmax[n])) bits |= (1u << n);
        }
    }
    pkb[t] = (uint16_t)bits;
}

// ------------------------------------------------------------- dilate -------
__global__ void dilate_kernel(const uint16_t* __restrict__ pkb, uint16_t* __restrict__ d5) {
    int t = blockIdx.x * 256 + threadIdx.x;
    if (t >= TPAD) return;
    int col = t % RPAD;
    uint32_t v = 0;
#pragma unroll
    for (int d = 0; d < 5; ++d)
        if (col + d < RPAD) v |= pkb[t + d];
    d5[t] = (uint16_t)v;
}

// -------------------------------------------------------------- mask --------
typedef __attribute__((ext_vector_type(4))) unsigned int v4u;
typedef __attribute__((ext_vector_type(8))) int          v8i;
typedef __attribute__((ext_vector_type(4))) int          v4i;
typedef __attribute__((ext_vector_type(2))) float        v2f;
typedef __attribute__((ext_vector_type(8))) float        v8f;

#define D5_BYTES  (TPAD * 2)            // 270848
#define CS_BYTES  (A_PAD * 8)           // 2944
#define SMEM_BYTES (D5_BYTES + CS_BYTES)

__global__ void mask_kernel(const uint16_t* __restrict__ d5g,
                            const uint16_t* __restrict__ pkg,
                            const float2*   __restrict__ csg,
                            float*          __restrict__ out) {
    extern __shared__ char smem[];
    uint16_t* s_d5 = (uint16_t*)smem;
    float2*   s_cs = (float2*)(smem + D5_BYTES);

    // small cos/sin table: plain cooperative copy
    for (int i = threadIdx.x; i < A_PAD; i += blockDim.x) s_cs[i] = csg[i];

    // 271KB D5 table: one TDM descriptor, issued by wave 0 (EXEC-independent DMA)
    if (threadIdx.x == 0) {
        uint32_t lds_off = (uint32_t)(uintptr_t)(void*)s_d5;   // low 32b of shared ptr = LDS byte offset
        uint64_t ga = (uint64_t)(uintptr_t)d5g;
        v4u g0 = {0u, 0u, 0u, 0u};
        g0.x = 1u;                                             // count = 1 valid descriptor
        g0.y = lds_off;                                        // lds_addr [63:32]
        g0.z = (uint32_t)(ga & 0xffffffffu);                   // global_addr[31:0]
        g0.w = (uint32_t)((ga >> 32) & 0x01ffffffu) | (2u << 30); // global_addr[56:32], type=2
        v8i g1 = {0, 0, 0, 0, 0, 0, 0, 0};
        g1.s0 = (1 << 16);                                     // data_size = 1 (2-byte elems)
        g1.s1 = (RPAD << 16);                                  // tensor_dim0[15:0]
        g1.s2 = ((A_PAD & 0xffff) << 16);                      // tensor_dim1[15:0]
        g1.s3 = (RPAD << 16);                                  // tile_dim0 = RPAD
        g1.s4 = A_PAD;                                         // tile_dim1 = 368, tile_dim2 = 0
        g1.s5 = RPAD;                                          // tensor_dim0_stride
        v4i g2 = {0, 0, 0, 0};                                 // 2D tensor: groups 2-4 unused
        v4i g3 = {0, 0, 0, 0};
        v8i g4 = {0, 0, 0, 0, 0, 0, 0, 0};
        __builtin_amdgcn_tensor_load_to_lds(g0, g1, g2, g3, g4, 0);
        __builtin_amdgcn_s_wait_tensorcnt(0);
    }
    __syncthreads();

    const int lane  = threadIdx.x & 31;
    const int wave  = threadIdx.x >> 5;
    const int laneN = lane & 15;             // pixel column within tile (B/C/D N index)
    const bool lo   = lane < 16;
    const int mAdd  = (lane & 16) >> 1;      // D rows: M = v (lanes 0-15), v+8 (lanes 16-31)

    // each wave: 4 pixel tiles of 16; block covers 8*64 = 512 pixels
    for (int pt = 0; pt < 4; ++pt) {
        const int p = blockIdx.x * 512 + wave * 64 + pt * 16 + laneN;
        const float x = (float)(p & 255) - 127.5f;
        const float y = (float)(p >> 8) - 127.5f;

        // B (4x16): K0=x, K1=y rows in lanes 0-15; K2=1, K3=0 rows in lanes 16-31
        v2f bfrag;
        bfrag.x = lo ? x : 1.0f;
        bfrag.y = lo ? y : 0.0f;

        uint32_t mask = 0;
        for (int at = 0; at < A_TILES; ++at) {
            const int aBase = at * 16;
            // A (16x4): rows (cos/dr, sin/dr, 180, 0); lanes 0-15 K0/K1, lanes 16-31 K2/K3
            float2 c = s_cs[aBase + laneN];
            v2f afrag;
            afrag.x = lo ? c.x * INV_DR : 180.0f;
            afrag.y = lo ? c.y * INV_DR : 0.0f;
            v8f acc = {0.f, 0.f, 0.f, 0.f, 0.f, 0.f, 0.f, 0.f};
            // f[a,p] tile = A x B : one matrix op replaces 512 scalar FMAs
            acc = __builtin_amdgcn_wmma_f32_16x16x4_f32(false, afrag, false, bfrag,
                                                        (short)0, acc, false, false);
            // 32-bit table indices (table is 271KB) -> SADDR + 32-bit voffset addressing
            const uint32_t rowBase = (uint32_t)((aBase + mAdd) * RPAD + 2);
#pragma unroll
            for (int v = 0; v < 8; ++v) {
                const float f = acc[v];
                const int rlo = (int)floorf(f - CW_F) + 1;     // first r in window
                const int rhi = (int)floorf(f + CW_F);         // last r in window
                const uint32_t oL = rowBase + (uint32_t)(v * RPAD) + (uint32_t)rlo;
                const uint32_t oG = rowBase + (uint32_t)(v * RPAD) + (uint32_t)rhi;
                mask |= (uint32_t)s_d5[oL] | (uint32_t)pkg[oG];
            }
        }
        // pixel N is held by lanes N and N+16: merge half-waves
        mask |= __shfl_xor((unsigned)mask, 16, 32);
        if (lo) {
#pragma unroll
            for (int n = 0; n < NMAPS; ++n)
                out[n * HW + p] = ((mask >> n) & 1u) ? 1.0f : 0.0f;
        }
    }
}

// ------------------------------------------------------------- launch -------
extern "C" void kernel_launch(void* const* d_in, const int* in_sizes, int n_in,
                              void* d_out, int out_size, void* d_ws, size_t ws_size,
                              hipStream_t stream) {
    const float* hough = (const float*)d_in[0];
    float* out = (float*)d_out;
    char* ws = (char*)d_ws;

    float*    gmax = (float*)(ws + WS_GMAX);
    float2*   cs   = (float2*)(ws + WS_CS);
    uint16_t* pkb  = (uint16_t*)(ws + WS_PKB);
    uint16_t* d5   = (uint16_t*)(ws + WS_D5);

    gmax_kernel<<<NMAPS, 256, 0, stream>>>(hough, gmax);
    trig_kernel<<<2, 256, 0, stream>>>(cs);
    peaks_kernel<<<(TPAD + 255) / 256, 256, 0, stream>>>(hough, gmax, pkb);
    dilate_kernel<<<(TPAD + 255) / 256, 256, 0, stream>>>(pkb, d5);
    mask_kernel<<<HW / 512, 256, SMEM_BYTES, stream>>>(d5, pkb, cs, out);
}